// DGGC_20452634264234
// MI455X (gfx1250) — compile-verified
//
#include <hip/hip_runtime.h>
#include <hip/hip_bf16.h>
#include <stddef.h>

// ---------------- problem constants ----------------
#define Bb   64
#define Nn   170
#define Tt   12
#define Cc   128
#define NP   176          // N padded to multiple of 16 for WMMA tiles
#define KTOP 136          // int(170 * 0.8)
#define BNT  (Bb*Nn*Tt)   // 130560 rows

typedef __attribute__((ext_vector_type(2))) float v2f;
typedef __attribute__((ext_vector_type(8))) float v8f;

__device__ __forceinline__ v8f wmma_f32_4(v2f a, v2f b, v8f c) {
  // D = A(16x4 f32) * B(4x16 f32) + C ; emits v_wmma_f32_16x16x4_f32
  return __builtin_amdgcn_wmma_f32_16x16x4_f32(false, a, false, b, (short)0, c,
                                               false, false);
}

// ---------------- kernel 0: zero-fill slack regions ----------------
__global__ void dggc_fillz_k(float* __restrict__ p, int n) {
  int i = blockIdx.x * 256 + threadIdx.x;
  if (i < n) p[i] = 0.0f;
}

// ---------------- kernel 1: conv 1x1 GEMM  H = X * Wc^T + bc ----------------
// X: [BNT, C], Wc: [C_out, C_in] row-major, H: [BNT, C]
__global__ void dggc_conv_gemm_k(const float* __restrict__ X,
                                 const float* __restrict__ W,
                                 const float* __restrict__ bias,
                                 float* __restrict__ H) {
  int lane = threadIdx.x & 31, wid = threadIdx.x >> 5;
  int tile = blockIdx.x * 8 + wid;      // = rowTile*8 + colTile
  int ct = tile & 7;
  size_t rowBase = (size_t)(tile >> 3) * 16;
  int lm = lane & 15;
  int kh = (lane >> 4) << 1;            // k-pair offset within K-step of 4
  const float* xr = X + (rowBase + lm) * Cc;              // A row m
  const float* wr = W + (size_t)(ct * 16 + lm) * Cc;      // B col n (=W row d)
  v8f acc = {0.f,0.f,0.f,0.f,0.f,0.f,0.f,0.f};
  for (int k = 0; k < Cc; k += 4) {
    v2f a = *(const v2f*)(xr + k + kh);       // A[m, k+kh], A[m, k+kh+1]
    v2f b = *(const v2f*)(wr + k + kh);       // B[k+kh, n], B[k+kh+1, n]
    acc = wmma_f32_4(a, b, acc);
  }
  int n = ct * 16 + lm;
  float bv = bias[n];
  int mh = (lane >> 4) * 8;
  float* o = H + (rowBase + mh) * Cc + n;
  for (int r = 0; r < 8; ++r) o[(size_t)r * Cc] = acc[r] + bv;
}

// ---------------- kernel 2: xs[b,n] = sum over (t, d) of H ----------------
__global__ void dggc_reduce_xs_k(const float* __restrict__ H,
                                 float* __restrict__ xs) {
  int bn = blockIdx.x;                        // b*N + n  (chunk is contiguous)
  const float* p = H + (size_t)bn * (Tt * Cc);
  float s = 0.f;
  for (int i = threadIdx.x; i < Tt * Cc; i += 256) s += p[i];
  __shared__ float red[256];
  red[threadIdx.x] = s; __syncthreads();
  for (int st = 128; st; st >>= 1) {
    if (threadIdx.x < st) red[threadIdx.x] += red[threadIdx.x + st];
    __syncthreads();
  }
  if (threadIdx.x == 0) xs[bn] = red[0];
}

// ---------------- kernel 3: S = relu(mem^T mem), diag=0.1 ----------------
__global__ void dggc_gram_k(const float* __restrict__ mem, float* __restrict__ S) {
  int idx = blockIdx.x * 256 + threadIdx.x;
  if (idx >= Nn * Nn) return;
  int n = idx / Nn, m = idx % Nn;
  float s = 0.f;
  for (int c = 0; c < Cc; ++c) s += mem[c * Nn + n] * mem[c * Nn + m];
  s = s > 0.f ? s : 0.f;
  if (n == m) s = 0.1f;
  S[idx] = s;
}

// ---------------- kernel 4: in-place row softmax (170 per row) ----------------
__global__ void dggc_rowsoftmax_k(float* __restrict__ S) {
  int n = blockIdx.x, tid = threadIdx.x;
  __shared__ float red[256];
  float* row = S + (size_t)n * Nn;
  float v = (tid < Nn) ? row[tid] : -3.4e38f;
  red[tid] = v; __syncthreads();
  for (int s = 128; s; s >>= 1) {
    if (tid < s) red[tid] = fmaxf(red[tid], red[tid + s]);
    __syncthreads();
  }
  float mx = red[0]; __syncthreads();
  float e = (tid < Nn) ? expf(v - mx) : 0.f;
  red[tid] = e; __syncthreads();
  for (int s = 128; s; s >>= 1) {
    if (tid < s) red[tid] += red[tid + s];
    __syncthreads();
  }
  float inv = 1.0f / red[0];
  if (tid < Nn) row[tid] = e * inv;
}

// ---------------- small NxN helpers for the power-series sum ----------------
__global__ void dggc_ident_k(float* __restrict__ Tm) {
  int idx = blockIdx.x * 256 + threadIdx.x;
  if (idx < Nn * Nn) Tm[idx] = (idx / Nn == idx % Nn) ? 1.f : 0.f;
}
__global__ void dggc_matadd_k(float* __restrict__ Tm, const float* __restrict__ U) {
  int idx = blockIdx.x * 256 + threadIdx.x;
  if (idx < Nn * Nn) Tm[idx] += U[idx];
}
__global__ void dggc_matnn_k(float* __restrict__ Cm, const float* __restrict__ A,
                             const float* __restrict__ B) {
  int idx = blockIdx.x * 256 + threadIdx.x;
  if (idx >= Nn * Nn) return;
  int i = idx / Nn, j = idx % Nn;
  const float* ar = A + (size_t)i * Nn;
  float s = 0.f;
  for (int k = 0; k < Nn; ++k) s += ar[k] * B[(size_t)k * Nn + j];
  Cm[idx] = s;
}

// ---- kernel 5: scores -> softmax -> softmax -> topk mask -> A_f^T (padded) ----
// one block per (b, n-padded); writes column n of AfT[b] (layout [m][n], ld=NP)
__global__ void dggc_build_aft_k(const float* __restrict__ xs,
                                 const float* __restrict__ Ss,
                                 float* __restrict__ AfT) {
  int b = blockIdx.x / NP, n = blockIdx.x % NP;
  int tid = threadIdx.x;
  float* col = AfT + (size_t)b * NP * NP + n;
  if (n >= Nn) {                       // zero pad columns
    if (tid < NP) col[(size_t)tid * NP] = 0.f;
    return;
  }
  __shared__ float row[Nn];
  __shared__ float red[256];
  float xv = xs[b * Nn + n] * 0.08838834764831845f;   // 1/sqrt(128)
  if (tid < Nn) {
    float s = xv * Ss[(size_t)n * Nn + tid];
    row[tid] = s > 0.f ? s : 0.f;
  }
  __syncthreads();
  for (int pass = 0; pass < 2; ++pass) {              // softmax twice
    float x = (tid < Nn) ? row[tid] : -3.4e38f;
    red[tid] = x; __syncthreads();
    for (int s = 128; s; s >>= 1) {
      if (tid < s) red[tid] = fmaxf(red[tid], red[tid + s]);
      __syncthreads();
    }
    float mx = red[0]; __syncthreads();
    float e = (tid < Nn) ? expf(x - mx) : 0.f;
    red[tid] = e; __syncthreads();
    for (int s = 128; s; s >>= 1) {
      if (tid < s) red[tid] += red[tid + s];
      __syncthreads();
    }
    float inv = 1.0f / red[0]; __syncthreads();
    if (tid < Nn) row[tid] = e * inv;
    __syncthreads();
  }
  // top-k keep mask (jax tie-break: lower index wins), write transposed
  if (tid < Nn) {
    float mv = row[tid];
    int cnt = 0;
    for (int j = 0; j < Nn; ++j) {
      float u = row[j];
      cnt += (u > mv) || (u == mv && j < tid);
    }
    col[(size_t)tid * NP] = (cnt < KTOP) ? mv : 0.f;
  } else if (tid < NP) {
    col[(size_t)tid * NP] = 0.f;       // zero pad rows
  }
}

// ---------------- kernel 6: diffusion  G[(b,m,t),c] = sum_n AfT[b][m][n]*In[(b,n,t),c]
__global__ void dggc_diff_gemm_k(const float* __restrict__ In,
                                 const float* __restrict__ AfT,
                                 float* __restrict__ Out) {
  int lane = threadIdx.x & 31, wid = threadIdx.x >> 5;
  int tile = blockIdx.x * 8 + wid;     // [bt(768)][mt(11)][ct(8)]
  int ct = tile & 7;
  int t2 = tile >> 3;
  int mt = t2 % 11;
  int bt = t2 / 11;
  int b = bt / Tt, t = bt % Tt;
  int lm = lane & 15, kh = (lane >> 4) << 1;
  const float* Ar = AfT + (size_t)b * NP * NP + (size_t)(mt * 16 + lm) * NP;
  const float* Ib = In + ((size_t)b * Nn * Tt + t) * Cc + (ct * 16 + lm);
  v8f acc = {0.f,0.f,0.f,0.f,0.f,0.f,0.f,0.f};
  for (int k = 0; k < NP; k += 4) {
    int k0 = k + kh;
    v2f a = *(const v2f*)(Ar + k0);              // A^T[m, k0], [m, k0+1]
    v2f bv;
    bv.x = Ib[(size_t)k0 * (Tt * Cc)];           // In[(b,k0,t), c]
    bv.y = Ib[(size_t)(k0 + 1) * (Tt * Cc)];     // zero-padded A kills slack reads
    acc = wmma_f32_4(a, bv, acc);
  }
  int c = ct * 16 + lm;
  int mb = mt * 16 + (lane >> 4) * 8;
  for (int r = 0; r < 8; ++r) {
    int m = mb + r;
    if (m < Nn) Out[((size_t)(b * Nn + m) * Tt + t) * Cc + c] = acc[r];
  }
}

// ---------------- kernel 7: gcn GEMM  out = [G1|G2] * W2^T + b2 ----------------
__global__ void dggc_gcn_gemm_k(const float* __restrict__ G1,
                                const float* __restrict__ G2,
                                const float* __restrict__ W2,  // [C, 2C]
                                const float* __restrict__ b2,
                                float* __restrict__ Outp) {
  int lane = threadIdx.x & 31, wid = threadIdx.x >> 5;
  int tile = blockIdx.x * 8 + wid;
  int ct = tile & 7;
  size_t rowBase = (size_t)(tile >> 3) * 16;
  int lm = lane & 15, kh = (lane >> 4) << 1;
  const float* g1 = G1 + (rowBase + lm) * Cc;
  const float* g2 = G2 + (rowBase + lm) * Cc;
  int n = ct * 16 + lm;
  const float* wr = W2 + (size_t)n * (2 * Cc);
  v8f acc = {0.f,0.f,0.f,0.f,0.f,0.f,0.f,0.f};
  for (int k = 0; k < Cc; k += 4) {
    v2f a = *(const v2f*)(g1 + k + kh);
    v2f b = *(const v2f*)(wr + k + kh);
    acc = wmma_f32_4(a, b, acc);
  }
  for (int k = 0; k < Cc; k += 4) {
    v2f a = *(const v2f*)(g2 + k + kh);
    v2f b = *(const v2f*)(wr + Cc + k + kh);
    acc = wmma_f32_4(a, b, acc);
  }
  float bv = b2[n];
  int mh = (lane >> 4) * 8;
  float* o = Outp + (rowBase + mh) * Cc + n;
  for (int r = 0; r < 8; ++r) o[(size_t)r * Cc] = acc[r] + bv;
}

// ---------------- host launcher ----------------
extern "C" void kernel_launch(void* const* d_in, const int* in_sizes, int n_in,
                              void* d_out, int out_size, void* d_ws, size_t ws_size,
                              hipStream_t stream) {
  const float* x     = (const float*)d_in[0];   // (B,N,T,C)
  const float* memry = (const float*)d_in[1];   // (C,N)
  const float* convw = (const float*)d_in[2];   // (C,C)
  const float* convb = (const float*)d_in[3];   // (C)
  const float* gcnw  = (const float*)d_in[4];   // (C,2C)
  const float* gcnb  = (const float*)d_in[5];   // (C)
  float* out = (float*)d_out;
  float* ws  = (float*)d_ws;

  const size_t BNTC = (size_t)BNT * Cc;   // 16,711,680
  const size_t SLK  = (size_t)72 * Cc;    // slack rows read via zero-padded A
  float* H   = ws;
  float* G1  = H  + BNTC + SLK;
  float* G2  = G1 + BNTC + SLK;
  float* AfT = G2 + BNTC + SLK;                  // B * 176 * 176
  float* xs  = AfT + (size_t)Bb * NP * NP;
  float* S   = xs + (size_t)Bb * Nn;             // S -> S_w (in place)
  float* Tm  = S  + (size_t)Nn * Nn;             // running power-series sum
  float* U1  = Tm + (size_t)Nn * Nn;
  float* PA  = U1 + (size_t)Nn * Nn;
  float* PB  = PA + (size_t)Nn * Nn;

  const int NN_BLOCKS = (Nn * Nn + 255) / 256;   // 113

  // zero slack so padded-K reads never inject NaN*0
  dggc_fillz_k<<<(int)((SLK + 255) / 256), 256, 0, stream>>>(H + BNTC, (int)SLK);
  dggc_fillz_k<<<(int)((SLK + 255) / 256), 256, 0, stream>>>(G1 + BNTC, (int)SLK);

  // 1x1 conv GEMM + xs reduction
  dggc_conv_gemm_k<<<BNT / 16, 256, 0, stream>>>(x, convw, convb, H);
  dggc_reduce_xs_k<<<Bb * Nn, 256, 0, stream>>>(H, xs);

  // adjacency: gram + softmax
  dggc_gram_k<<<NN_BLOCKS, 256, 0, stream>>>(memry, S);
  dggc_rowsoftmax_k<<<Nn, 256, 0, stream>>>(S);

  // Ss = sum_{k=0}^{170} S_w^k via binary doubling:
  // state (T=F(e), P=S^{e+1}); D: e->2e+1 (T+=P*T, P=P*P); I: e->e+1 (T+=P, P=P*S)
  // sequence reaching e=170:
  dggc_ident_k<<<NN_BLOCKS, 256, 0, stream>>>(Tm);
  float* P = S;                  // P starts as S_w (kept intact; never overwritten)
  float* pb[2] = {PA, PB};
  int pi = 0;
  const char* ops = "DDIDDIDDIDI";
  for (const char* o = ops; *o; ++o) {
    if (*o == 'D') {
      dggc_matnn_k<<<NN_BLOCKS, 256, 0, stream>>>(U1, P, Tm);
      dggc_matadd_k<<<NN_BLOCKS, 256, 0, stream>>>(Tm, U1);
      float* Pn = pb[pi]; pi ^= 1;
      dggc_matnn_k<<<NN_BLOCKS, 256, 0, stream>>>(Pn, P, P);
      P = Pn;
    } else {
      dggc_matadd_k<<<NN_BLOCKS, 256, 0, stream>>>(Tm, P);
      float* Pn = pb[pi]; pi ^= 1;
      dggc_matnn_k<<<NN_BLOCKS, 256, 0, stream>>>(Pn, P, S);
      P = Pn;
    }
  }

  // per-batch adjacency: relu/softmax^2/topk -> transposed padded A_f
  dggc_build_aft_k<<<Bb * NP, 256, 0, stream>>>(xs, Tm, AfT);

  // two diffusion steps (WMMA per (b,t) slice)
  dggc_diff_gemm_k<<<(Bb * Tt * 11 * 8) / 8, 256, 0, stream>>>(H, AfT, G1);
  dggc_diff_gemm_k<<<(Bb * Tt * 11 * 8) / 8, 256, 0, stream>>>(G1, AfT, G2);

  // output GEMM straight into d_out ([b,n,t,c] == [row,d])
  dggc_gcn_gemm_k<<<BNT / 16, 256, 0, stream>>>(G1, G2, gcnw, gcnb, out);
}